// DilatedNeighborhoodAttention1D_46170898432358
// MI455X (gfx1250) — compile-verified
//
#include <hip/hip_runtime.h>

// CDNA5 / gfx1250, wave32. Full-fp32 attention using V_WMMA_F32_16X16X4_F32,
// software-pipelined loads, and async global->LDS staging for V tiles.

typedef float v2f __attribute__((ext_vector_type(2)));
typedef float v8f __attribute__((ext_vector_type(8)));
typedef int   v4i __attribute__((ext_vector_type(4)));

#define WMMA_F32(a, b, c) \
  __builtin_amdgcn_wmma_f32_16x16x4_f32(false, (a), false, (b), (short)0, (c), false, false)

// ---- CDNA5 async global->LDS (ASYNCcnt) path, guarded so a builtin-name
//      mismatch falls back to direct vector loads instead of breaking the build.
#if defined(__has_builtin)
#if __has_builtin(__builtin_amdgcn_global_load_async_to_lds_b128) && \
    __has_builtin(__builtin_amdgcn_s_wait_asynccnt)
#define HAVE_ASYNC_LDS 1
#endif
#endif
#ifndef HAVE_ASYNC_LDS
#define HAVE_ASYNC_LDS 0
#endif

#if HAVE_ASYNC_LDS
// Builtin signature (from the round-2 diagnostic): param0 = int4 AS(1)*,
// param1 = int4 AS(3)*, then two immediates (offset, cpol).
#define ASYNC_COPY_B128(gsrc, ldst)                                   \
  __builtin_amdgcn_global_load_async_to_lds_b128(                     \
      (__attribute__((address_space(1))) v4i*)(gsrc),                 \
      (__attribute__((address_space(3))) v4i*)(ldst), 0, 0)
#define WAIT_ASYNC0()                          \
  do {                                         \
    __builtin_amdgcn_s_wait_asynccnt(0);       \
    asm volatile("" ::: "memory");             \
  } while (0)
#endif

constexpr int   kB   = 4;
constexpr int   kN   = 2048;
constexpr int   kC   = 256;
constexpr int   kH   = 8;
constexpr int   kHd  = 32;
constexpr float kScale = 0.17677669529663687f;  // 1/sqrt(32)
constexpr int   kVStride = 48;                  // LDS row stride (floats) for V tile
constexpr int   kVBuf    = 16 * kVStride;       // one V tile buffer (floats)

// ---------------------------------------------------------------------------
// Kernel 1: QKV projection.  out(8192x768) = x(8192x256) @ qkv_w^T + qkv_b,
// scattered into q/k/v buffers laid out (B,H,N,hd).
// One wave computes a 16(M) x 64(O) tile; K stepped by 4 (f32 WMMA) with
// register double-buffering of the A/B fragments.
// ---------------------------------------------------------------------------
__global__ __launch_bounds__(128) void qkv_proj_kernel(
    const float* __restrict__ x, const float* __restrict__ w,
    const float* __restrict__ bias, float* __restrict__ qb,
    float* __restrict__ kb, float* __restrict__ vb) {
  const int lane = threadIdx.x & 31;
  const int half = lane >> 4;
  const int lq   = lane & 15;
  const int wave = blockIdx.x * 4 + (threadIdx.x >> 5);
  const int mt   = wave / 12;   // 512 row tiles of 16
  const int ot   = wave % 12;   // 12 col tiles of 64 (768 outputs)

  v8f acc[4];
#pragma unroll
  for (int t = 0; t < 4; ++t)
#pragma unroll
    for (int r = 0; r < 8; ++r) acc[t][r] = 0.0f;

  const float* xrow  = x + (size_t)(mt * 16 + lq) * kC + half * 2;
  const float* wrow  = w + (size_t)(ot * 64 + lq) * kC + half * 2;  // +t*16*kC per t

  v2f a_cur = *(const v2f*)(xrow);
  v2f b_cur[4];
#pragma unroll
  for (int t = 0; t < 4; ++t) b_cur[t] = *(const v2f*)(wrow + (size_t)t * 16 * kC);

  for (int k0 = 0; k0 < kC - 4; k0 += 4) {
    // prefetch next fragments first so current WMMAs overlap the loads
    v2f a_nxt = *(const v2f*)(xrow + k0 + 4);
    v2f b_nxt[4];
#pragma unroll
    for (int t = 0; t < 4; ++t)
      b_nxt[t] = *(const v2f*)(wrow + (size_t)t * 16 * kC + k0 + 4);
#pragma unroll
    for (int t = 0; t < 4; ++t) acc[t] = WMMA_F32(a_cur, b_cur[t], acc[t]);
    a_cur = a_nxt;
#pragma unroll
    for (int t = 0; t < 4; ++t) b_cur[t] = b_nxt[t];
  }
#pragma unroll
  for (int t = 0; t < 4; ++t) acc[t] = WMMA_F32(a_cur, b_cur[t], acc[t]);

#pragma unroll
  for (int t = 0; t < 4; ++t) {
    const int o   = ot * 64 + t * 16 + lq;   // 0..767
    const int sel = o >> 8;                  // 0=q 1=k 2=v (uniform per t)
    const int h   = (o >> 5) & 7;
    const int d   = o & 31;
    float* dst = (sel == 0) ? qb : ((sel == 1) ? kb : vb);
    const float bv = bias[o];
#pragma unroll
    for (int r = 0; r < 8; ++r) {
      const int row = mt * 16 + r + 8 * half;  // b*2048 + n
      const int b   = row >> 11;
      const int n   = row & (kN - 1);
      dst[(((size_t)(b * kH + h) * kN) + n) * kHd + d] = acc[t][r] + bv;
    }
  }
}

// ---------------------------------------------------------------------------
// Kernel 2: flash attention.  One wave = one (b,h) x 16-query tile; streams
// all 2048 keys in 16-wide tiles with online softmax.
//  - S and O tiles use the WMMA C/D layout: each row's 16 columns live in one
//    16-lane half, so row stats come from __shfl_xor(1/2/4/8) and apply
//    directly (per accumulator VGPR) to the O tiles.
//  - K fragments are register double-buffered (prefetch next tile while the
//    softmax VALU phase runs).
//  - V tiles are staged to LDS with double-buffered async b128 copies
//    (ASYNCcnt), waited one iteration later; row stride 48 floats keeps the
//    b128 writes 16B-aligned and the two half-wave read groups bank-disjoint.
//  - P is transposed through a 16x18 LDS tile (conflict-free, 8B aligned).
// ---------------------------------------------------------------------------
__global__ __launch_bounds__(128) void flash_attn_kernel(
    const float* __restrict__ q, const float* __restrict__ k,
    const float* __restrict__ v, const float* __restrict__ table,
    float* __restrict__ out) {
  __shared__ float pshm_all[4][16 * 18];
#if HAVE_ASYNC_LDS
  __shared__ float vshm_all[4][2 * kVBuf];
#endif
  __shared__ float tbl[16];
  if (threadIdx.x < 13) tbl[threadIdx.x] = table[threadIdx.x];
  __syncthreads();

  const int lane = threadIdx.x & 31;
  const int half = lane >> 4;
  const int lq   = lane & 15;
  const int wid  = threadIdx.x >> 5;
  const int wave = blockIdx.x * 4 + wid;
  const int bh   = wave >> 7;   // b*8 + h  (0..31)
  const int qt   = wave & 127;  // query tile (0..127)
  float* pshm = pshm_all[wid];
#if HAVE_ASYNC_LDS
  float* vshm = vshm_all[wid];
#endif

  // Resident Q fragments (A layout): row qt*16+lq, d = ks*4 + half*2 + {0,1}.
  v2f qf[8];
  const float* qp = q + (size_t)(bh * kN + qt * 16 + lq) * kHd + half * 2;
#pragma unroll
  for (int ks = 0; ks < 8; ++ks) qf[ks] = *(const v2f*)(qp + ks * 4);

  v8f olo, ohi;
  float mrun[8], lrun[8];
  int qmod[8];
#pragma unroll
  for (int r = 0; r < 8; ++r) {
    olo[r] = 0.0f; ohi[r] = 0.0f;
    mrun[r] = -1.0e30f; lrun[r] = 0.0f;
    qmod[r] = (qt * 16 + r + 8 * half) % 7;
  }

  const float* kbase = k + (size_t)(bh * kN + lq) * kHd + half * 2;  // + j*kHd
  const float* vbase = v + (size_t)bh * kN * kHd;                    // + j*kHd

  // Preload K fragments for j=0.
  v2f kf[8];
#pragma unroll
  for (int ks = 0; ks < 8; ++ks) kf[ks] = *(const v2f*)(kbase + ks * 4);

#if HAVE_ASYNC_LDS
  // Kick off async copy of V tile j=0 into buffer 0.
  {
    const float* vsrc = vbase;
#pragma unroll
    for (int i = 0; i < 4; ++i) {
      const int idx = i * 32 + lane;
      const int row = idx >> 3;
      const int col = (idx & 7) * 4;  // float index within row
      ASYNC_COPY_B128(vsrc + row * kHd + col, vshm + row * kVStride + col);
    }
  }
#endif

  for (int j = 0; j < kN; j += 16) {
    // ---- S = Q K^T (16x16), 8 WMMAs over hd=32 ----
    v8f s;
#pragma unroll
    for (int r = 0; r < 8; ++r) s[r] = 0.0f;
#pragma unroll
    for (int ks = 0; ks < 8; ++ks) s = WMMA_F32(qf[ks], kf[ks], s);

    // ---- prefetch K fragments for next tile (overlaps softmax phase) ----
    if (j + 16 < kN) {
      const float* kp = kbase + (size_t)(j + 16) * kHd;
#pragma unroll
      for (int ks = 0; ks < 8; ++ks) kf[ks] = *(const v2f*)(kp + ks * 4);
    }

#if HAVE_ASYNC_LDS
    const int cur = (j >> 4) & 1;
    // Current V tile is ready once its async copies (issued last iteration)
    // have drained; then immediately launch next tile into the other buffer.
    WAIT_ASYNC0();
    if (j + 16 < kN) {
      const float* vsrc = vbase + (size_t)(j + 16) * kHd;
      float* vdst = vshm + (cur ^ 1) * kVBuf;
#pragma unroll
      for (int i = 0; i < 4; ++i) {
        const int idx = i * 32 + lane;
        const int row = idx >> 3;
        const int col = (idx & 7) * 4;
        ASYNC_COPY_B128(vsrc + row * kHd + col, vdst + row * kVStride + col);
      }
    }
    const float* vt = vshm + cur * kVBuf;       // LDS, row stride kVStride
#else
    const float* vt = vbase + (size_t)j * kHd;  // global, row stride kHd
#endif

    // ---- scale + periodic relative bias, online softmax ----
    const int cm = (j + lq) % 7;             // key-column mod 7 (per lane)
#pragma unroll
    for (int r = 0; r < 8; ++r) {
      float val = s[r] * kScale + tbl[qmod[r] - cm + 6];
      float t = val;                          // row max over 16 lanes (half)
      t = fmaxf(t, __shfl_xor(t, 1, 32));
      t = fmaxf(t, __shfl_xor(t, 2, 32));
      t = fmaxf(t, __shfl_xor(t, 4, 32));
      t = fmaxf(t, __shfl_xor(t, 8, 32));
      const float mnew = fmaxf(mrun[r], t);
      const float fac  = __expf(mrun[r] - mnew);
      const float p    = __expf(val - mnew);
      float su = p;                           // row sum
      su += __shfl_xor(su, 1, 32);
      su += __shfl_xor(su, 2, 32);
      su += __shfl_xor(su, 4, 32);
      su += __shfl_xor(su, 8, 32);
      lrun[r] = lrun[r] * fac + su;
      mrun[r] = mnew;
      olo[r] *= fac;
      ohi[r] *= fac;
      pshm[(r + 8 * half) * 18 + lq] = p;     // stage P for transpose
    }

    // ---- O += P V : P re-read in A layout, V as B fragments ----
#if HAVE_ASYNC_LDS
    constexpr int vs = kVStride;
#else
    constexpr int vs = kHd;
#endif
#pragma unroll
    for (int ks2 = 0; ks2 < 4; ++ks2) {
      v2f pf = *(const v2f*)(pshm + lq * 18 + ks2 * 4 + half * 2);
      const int key = ks2 * 4 + half * 2;
      v2f vlo, vhi;
      vlo.x = vt[(key + 0) * vs + lq];
      vlo.y = vt[(key + 1) * vs + lq];
      vhi.x = vt[(key + 0) * vs + 16 + lq];
      vhi.y = vt[(key + 1) * vs + 16 + lq];
      olo = WMMA_F32(pf, vlo, olo);
      ohi = WMMA_F32(pf, vhi, ohi);
    }
  }

  // ---- normalize + write (B,N,C) pre-projection activations ----
  const int b = bh >> 3, h = bh & 7;
#pragma unroll
  for (int r = 0; r < 8; ++r) {
    const float inv = 1.0f / lrun[r];
    const size_t row = (size_t)(b * kN + qt * 16 + r + 8 * half);
    out[row * kC + h * kHd + lq]      = olo[r] * inv;
    out[row * kC + h * kHd + 16 + lq] = ohi[r] * inv;
  }
}

// ---------------------------------------------------------------------------
// Kernel 3: output projection.  d_out(8192x256) = attn @ proj_w^T + proj_b.
// Same pipelined 16x64-tile WMMA GEMM.
// ---------------------------------------------------------------------------
__global__ __launch_bounds__(128) void out_proj_kernel(
    const float* __restrict__ a, const float* __restrict__ w,
    const float* __restrict__ bias, float* __restrict__ out) {
  const int lane = threadIdx.x & 31;
  const int half = lane >> 4;
  const int lq   = lane & 15;
  const int wave = blockIdx.x * 4 + (threadIdx.x >> 5);
  const int mt   = wave >> 2;   // 512 row tiles
  const int ot   = wave & 3;    // 4 col tiles of 64

  v8f acc[4];
#pragma unroll
  for (int t = 0; t < 4; ++t)
#pragma unroll
    for (int r = 0; r < 8; ++r) acc[t][r] = 0.0f;

  const float* arow = a + (size_t)(mt * 16 + lq) * kC + half * 2;
  const float* wrow = w + (size_t)(ot * 64 + lq) * kC + half * 2;

  v2f a_cur = *(const v2f*)(arow);
  v2f b_cur[4];
#pragma unroll
  for (int t = 0; t < 4; ++t) b_cur[t] = *(const v2f*)(wrow + (size_t)t * 16 * kC);

  for (int k0 = 0; k0 < kC - 4; k0 += 4) {
    v2f a_nxt = *(const v2f*)(arow + k0 + 4);
    v2f b_nxt[4];
#pragma unroll
    for (int t = 0; t < 4; ++t)
      b_nxt[t] = *(const v2f*)(wrow + (size_t)t * 16 * kC + k0 + 4);
#pragma unroll
    for (int t = 0; t < 4; ++t) acc[t] = WMMA_F32(a_cur, b_cur[t], acc[t]);
    a_cur = a_nxt;
#pragma unroll
    for (int t = 0; t < 4; ++t) b_cur[t] = b_nxt[t];
  }
#pragma unroll
  for (int t = 0; t < 4; ++t) acc[t] = WMMA_F32(a_cur, b_cur[t], acc[t]);

#pragma unroll
  for (int t = 0; t < 4; ++t) {
    const int o = ot * 64 + t * 16 + lq;
    const float bv = bias[o];
#pragma unroll
    for (int r = 0; r < 8; ++r) {
      const int row = mt * 16 + r + 8 * half;
      out[(size_t)row * kC + o] = acc[t][r] + bv;
    }
  }
}

// ---------------------------------------------------------------------------
extern "C" void kernel_launch(void* const* d_in, const int* in_sizes, int n_in,
                              void* d_out, int out_size, void* d_ws, size_t ws_size,
                              hipStream_t stream) {
  (void)in_sizes; (void)n_in; (void)out_size; (void)ws_size;
  const float* x       = (const float*)d_in[0];
  const float* qkv_w   = (const float*)d_in[1];
  const float* qkv_b   = (const float*)d_in[2];
  const float* proj_w  = (const float*)d_in[3];
  const float* proj_b  = (const float*)d_in[4];
  const float* table   = (const float*)d_in[5];

  float* ws = (float*)d_ws;
  const size_t perBuf = (size_t)kB * kH * kN * kHd;  // 2,097,152 floats (8 MB)
  float* qb   = ws;
  float* kb   = ws + perBuf;
  float* vb   = ws + 2 * perBuf;
  float* attn = ws + 3 * perBuf;                     // 8192*256 floats (8 MB)

  // 512*12 = 6144 waves, 4 waves / block
  qkv_proj_kernel<<<1536, 128, 0, stream>>>(x, qkv_w, qkv_b, qb, kb, vb);
  // 4*8*128 = 4096 waves
  flash_attn_kernel<<<1024, 128, 0, stream>>>(qb, kb, vb, table, attn);
  // 512*4 = 2048 waves
  out_proj_kernel<<<512, 128, 0, stream>>>(attn, proj_w, proj_b, (float*)d_out);
}